// JumpResGMEmbedder_15178414424419
// MI455X (gfx1250) — compile-verified
//
#include <hip/hip_runtime.h>

// ---------------------------------------------------------------------------
// JumpResGMEmbedder on MI455X (gfx1250).
// Memory-bound graph conv + GraphNorm + DeepSets readout.
// GEMMs run on V_WMMA_F32_16X16X4_F32 (full fp32 precision; compute is far
// from the bottleneck so no precision downgrade is warranted).
// ---------------------------------------------------------------------------

typedef __attribute__((ext_vector_type(2))) float v2f;
typedef __attribute__((ext_vector_type(8))) float v8f;

#define DD 64
#define N_LAYERS 3
#define EPSV 1e-5f
#define SLOPEV 0.01f

__device__ __forceinline__ float lrelu_f(float x) { return x > 0.0f ? x : SLOPEV * x; }

// ---------------------------------------------------------------- utilities
__global__ void k_zero(float* __restrict__ p, long n) {
  long i = (long)blockIdx.x * blockDim.x + threadIdx.x;
  if (i < n) p[i] = 0.0f;
}

__global__ void k_deg(const int* __restrict__ src, const int* __restrict__ dst,
                      float* __restrict__ deg_out, float* __restrict__ deg_in, int E) {
  int i = blockIdx.x * blockDim.x + threadIdx.x;
  if (i < E) {
    atomicAdd(&deg_out[src[i]], 1.0f);
    atomicAdd(&deg_in[dst[i]], 1.0f);
  }
}

__global__ void k_rsqrt_deg(float* __restrict__ p, long n) {
  long i = (long)blockIdx.x * blockDim.x + threadIdx.x;
  if (i < n) {
    float d = p[i];
    d = d < 1.0f ? 1.0f : d;
    p[i] = rsqrtf(d);
  }
}

// ---------------------------------------------------------------- WMMA GEMM
// out[M,64] = act((x .* rowscale) @ W + bias); optional column-sum accumulate.
// One wave per 16-row tile, 4x 16x16 f32 accumulators, K=64 via 16 steps of
// V_WMMA_F32_16X16X4_F32. W and the A tile are staged in padded LDS.
__global__ __launch_bounds__(256)
void k_gemm64(const float* __restrict__ x, const float* __restrict__ rowscale,
              const float* __restrict__ W, const float* __restrict__ bias,
              float* __restrict__ out, float* __restrict__ colsum,
              int relu_flag, int ntiles, long Nrows) {
  __shared__ __align__(16) float Ws[64 * 65];      // 64x64 weights, stride 65
  __shared__ __align__(16) float As[8][16 * 66];   // per-wave 16x64 A tile, stride 66
  const int tid = threadIdx.x;

  // cooperative load of W into LDS (coalesced global reads)
  #pragma unroll
  for (int j = 0; j < 16; ++j) {
    int i = tid + j * 256;
    Ws[(i >> 6) * 65 + (i & 63)] = W[i];
  }
  __syncthreads();

  const int wv = tid >> 5;          // wave32: 8 waves per block
  const int ln = tid & 31;
  const int tile = blockIdx.x * 8 + wv;
  if (tile >= ntiles) return;

  const long rowbase = (long)tile * 16;

  // load 16x64 A tile (row-scaled) into this wave's LDS slab
  #pragma unroll 4
  for (int j = 0; j < 32; ++j) {
    int i = ln + j * 32;
    int r = i >> 6, c = i & 63;
    long rr = rowbase + r;
    float v = 0.0f;
    if (rr < Nrows) {
      v = x[rr * DD + c];
      if (rowscale) v *= rowscale[rr];
    }
    As[wv][r * 66 + c] = v;
  }

  // A 16x4 layout: lane L -> row M=L&15, K pair (kk + 2*(L>>4)) .. +1
  // B 4x16 layout: lane L -> col N=L&15, K pair (kk + 2*(L>>4)) .. +1
  const int m  = ln & 15;
  const int hi = ln >> 4;
  const int kb = hi * 2;

  v8f acc0 = {}, acc1 = {}, acc2 = {}, acc3 = {};
  #pragma unroll
  for (int kk = 0; kk < 64; kk += 4) {
    v2f a;
    {
      const float2 t = *(const float2*)&As[wv][m * 66 + kk + kb];
      a.x = t.x; a.y = t.y;
    }
    const int r0 = (kk + kb) * 65, r1 = (kk + kb + 1) * 65;
    v2f b0, b1, b2, b3;
    b0.x = Ws[r0 + m +  0]; b0.y = Ws[r1 + m +  0];
    b1.x = Ws[r0 + m + 16]; b1.y = Ws[r1 + m + 16];
    b2.x = Ws[r0 + m + 32]; b2.y = Ws[r1 + m + 32];
    b3.x = Ws[r0 + m + 48]; b3.y = Ws[r1 + m + 48];
    acc0 = __builtin_amdgcn_wmma_f32_16x16x4_f32(false, a, false, b0, (short)0, acc0, false, false);
    acc1 = __builtin_amdgcn_wmma_f32_16x16x4_f32(false, a, false, b1, (short)0, acc1, false, false);
    acc2 = __builtin_amdgcn_wmma_f32_16x16x4_f32(false, a, false, b2, (short)0, acc2, false, false);
    acc3 = __builtin_amdgcn_wmma_f32_16x16x4_f32(false, a, false, b3, (short)0, acc3, false, false);
  }

  // epilogue: VGPR r of C holds row (r + 8*hi), column nt*16 + (ln&15)
  v8f accs[4] = {acc0, acc1, acc2, acc3};
  #pragma unroll
  for (int nt = 0; nt < 4; ++nt) {
    const int col = nt * 16 + m;
    const float bv = bias ? bias[col] : 0.0f;
    float csum = 0.0f;
    #pragma unroll
    for (int r = 0; r < 8; ++r) {
      long row = rowbase + r + 8 * hi;
      float v = accs[nt][r] + bv;
      if (relu_flag) v = v > 0.0f ? v : 0.0f;
      if (row < Nrows) {
        if (out) out[row * DD + col] = v;
        csum += v;
      }
    }
    if (colsum) atomicAdd(&colsum[col], csum);
  }
}

// ---------------------------------------------------------------- scatter
__global__ void k_scatter(const float* __restrict__ h, const int* __restrict__ src,
                          const int* __restrict__ dst, float* __restrict__ agg, int E) {
  long i = (long)blockIdx.x * blockDim.x + threadIdx.x;
  int e = (int)(i >> 6);
  int d = (int)(i & 63);
  if (e < E) {
    int s = src[e], t = dst[e];
    atomicAdd(&agg[(long)t * DD + d], h[(long)s * DD + d]);
  }
}

// ------------------------------------------------------- column reductions
// sq==0: outsum[d] += sum_r agg[r,d]*norm_in[r]
// sq==1: outsum[d] += sum_r (agg[r,d]*norm_in[r] - amu[d])^2
__global__ __launch_bounds__(256)
void k_colsum_conv(const float* __restrict__ agg, const float* __restrict__ norm_in,
                   const float* __restrict__ amu, float* __restrict__ outsum,
                   long N, int sq) {
  __shared__ float red[256];
  const int tid = threadIdx.x;
  const int d = tid & 63;
  const int slot = tid >> 6;
  const float mu = sq ? amu[d] : 0.0f;
  float acc = 0.0f;
  for (long r = (long)blockIdx.x * 4 + slot; r < N; r += (long)gridDim.x * 4) {
    float v = agg[r * DD + d] * norm_in[r];
    if (sq) { float s = v - mu; acc += s * s; }
    else acc += v;
  }
  red[tid] = acc;
  __syncthreads();
  if (tid < 64)
    atomicAdd(&outsum[tid], red[tid] + red[tid + 64] + red[tid + 128] + red[tid + 192]);
}

__global__ void k_amu(const float* __restrict__ colsum, const float* __restrict__ alpha,
                      float* __restrict__ amu, float invN) {
  int d = threadIdx.x;
  amu[d] = alpha[d] * colsum[d] * invN;
}

__global__ void k_rstd(const float* __restrict__ sumsq, float* __restrict__ rstd, float invN) {
  int d = threadIdx.x;
  rstd[d] = rsqrtf(sumsq[d] * invN + EPSV);
}

// --------------------------------------------- GraphNorm + lrelu + residual
__global__ void k_finalize(const float* __restrict__ agg, const float* __restrict__ norm_in,
                           const float* __restrict__ amu, const float* __restrict__ rstd,
                           const float* __restrict__ gamma, const float* __restrict__ beta,
                           const float* __restrict__ res, float* __restrict__ u_out,
                           float* __restrict__ y_out, long total) {
  long i = (long)blockIdx.x * blockDim.x + threadIdx.x;
  if (i >= total) return;
  int d = (int)(i & 63);
  long r = i >> 6;
  float conv = agg[i] * norm_in[r];
  float s = conv - amu[d];
  float u = lrelu_f(gamma[d] * s * rstd[d] + beta[d]);
  u_out[i] = u;
  y_out[i] = u + res[i];
}

// ----------------------------------------------------------- readout GEMV
__global__ void k_readout(const float* __restrict__ rosum, const float* __restrict__ W2,
                          const float* __restrict__ b2, float* __restrict__ out, float invN) {
  int j = threadIdx.x; // 64 threads
  float acc = b2[j];
  #pragma unroll 8
  for (int k = 0; k < 64; ++k) acc += rosum[k] * invN * W2[k * DD + j];
  out[j] = lrelu_f(acc);
}

// ---------------------------------------------------------------------------
extern "C" void kernel_launch(void* const* d_in, const int* in_sizes, int n_in,
                              void* d_out, int out_size, void* d_ws, size_t ws_size,
                              hipStream_t stream) {
  const float* x0    = (const float*)d_in[0];
  const int*   src   = (const int*)d_in[1];
  const int*   dst   = (const int*)d_in[2];
  const float* Wc    = (const float*)d_in[3];
  const float* alpha = (const float*)d_in[4];
  const float* gamma = (const float*)d_in[5];
  const float* beta  = (const float*)d_in[6];
  const float* W1    = (const float*)d_in[7];
  const float* b1    = (const float*)d_in[8];
  const float* W2    = (const float*)d_in[9];
  const float* b2    = (const float*)d_in[10];
  float* out = (float*)d_out;

  const long N = in_sizes[0] / DD;
  const int  E = in_sizes[1];

  float* ws = (float*)d_ws;
  float* norm_out = ws;  ws += N;       // degree -> rsqrt
  float* norm_in  = ws;  ws += N;       // contiguous with norm_out
  float* h    = ws;      ws += N * DD;  // message matrix
  float* agg  = ws;      ws += N * DD;  // scatter destination
  float* y    = ws;      ws += N * DD;  // residual-added output / next conv input
  float* ua   = ws;      ws += N * DD;  // u ping
  float* ub   = ws;      ws += N * DD;  // u pong
  float* colsum = ws;    ws += 64;      // colsum, sumsq, rosum contiguous
  float* sumsq  = ws;    ws += 64;
  float* rosum  = ws;    ws += 64;
  float* amu    = ws;    ws += 64;
  float* rstd   = ws;    ws += 64;

  const float invN = 1.0f / (float)N;
  const long twoN = 2 * N;
  const long ND   = N * DD;
  const int  ntiles = (int)((N + 15) / 16);

  // degrees -> symmetric norms
  k_zero<<<(int)((twoN + 255) / 256), 256, 0, stream>>>(norm_out, twoN);
  k_deg<<<(E + 255) / 256, 256, 0, stream>>>(src, dst, norm_out, norm_in, E);
  k_rsqrt_deg<<<(int)((twoN + 255) / 256), 256, 0, stream>>>(norm_out, twoN);

  const float* xcur = x0;
  float* ucur = ua;
  float* uprev = ub;

  for (int L = 0; L < N_LAYERS; ++L) {
    const float* WcL = Wc + (long)L * DD * DD;

    // h = (xcur .* norm_out) @ Wc[L]
    k_gemm64<<<(ntiles + 7) / 8, 256, 0, stream>>>(
        xcur, norm_out, WcL, nullptr, h, nullptr, 0, ntiles, N);

    // agg = segment_sum(h[src] -> dst)
    k_zero<<<(int)((ND + 255) / 256), 256, 0, stream>>>(agg, ND);
    {
      long tot = (long)E * DD;
      k_scatter<<<(int)((tot + 255) / 256), 256, 0, stream>>>(h, src, dst, agg, E);
    }

    // GraphNorm statistics over conv = agg .* norm_in
    k_zero<<<1, 256, 0, stream>>>(colsum, 192); // colsum + sumsq + rosum
    k_colsum_conv<<<1024, 256, 0, stream>>>(agg, norm_in, nullptr, colsum, N, 0);
    k_amu<<<1, 64, 0, stream>>>(colsum, alpha + L * DD, amu, invN);
    k_colsum_conv<<<1024, 256, 0, stream>>>(agg, norm_in, amu, sumsq, N, 1);
    k_rstd<<<1, 64, 0, stream>>>(sumsq, rstd, invN);

    // u = lrelu(gnorm(conv)); y = u + residual
    const float* res = (L == 0) ? x0 : (const float*)uprev;
    k_finalize<<<(int)((ND + 255) / 256), 256, 0, stream>>>(
        agg, norm_in, amu, rstd, gamma + L * DD, beta + L * DD, res, ucur, y, ND);

    // readout: colsum of relu(x_ro @ W1 + b1) accumulated in the GEMM epilogue
    const float* rox = (L == 0) ? (const float*)ucur : (const float*)y;
    k_gemm64<<<(ntiles + 7) / 8, 256, 0, stream>>>(
        rox, nullptr, W1 + (long)L * DD * DD, b1 + L * DD, nullptr, rosum, 1, ntiles, N);
    k_readout<<<1, 64, 0, stream>>>(rosum, W2 + (long)L * DD * DD, b2 + L * DD,
                                    out + L * DD, invN);

    // rotate buffers for next layer
    xcur = y;
    float* tmp = uprev; uprev = ucur; ucur = tmp;
  }
}